// OptNetLatent_54606214201656
// MI455X (gfx1250) — compile-verified
//
#include <hip/hip_runtime.h>
#include <math.h>

// Problem constants (match reference)
#define Bn   1024
#define DIN  128
#define Nn   64
#define Mm   64
#define Qq   16
#define DOUT 64
#define EPSQ 1e-4f
#define NIT  25
#define SIG  0.1f

typedef __attribute__((ext_vector_type(2))) float v2f;
typedef __attribute__((ext_vector_type(8))) float v8f;

// One workgroup (128 threads = 4 wave32) per batch element.
// All per-batch state lives in LDS; H-formation uses V_WMMA_F32_16X16X4_F32.
__global__ __launch_bounds__(128)
void optnet_ipm_kernel(const float* __restrict__ x,   const float* __restrict__ Win,
                       const float* __restrict__ bin, const float* __restrict__ Lr,
                       const float* __restrict__ Gg,  const float* __restrict__ zp,
                       const float* __restrict__ sp,  const float* __restrict__ Aa,
                       const float* __restrict__ beq, const float* __restrict__ Wout,
                       const float* __restrict__ bout, float* __restrict__ out)
{
    __shared__ float Qs[Nn][Nn];        // Q = tril(L) tril(L)^T + eps I
    __shared__ float Gs[Mm][Nn];        // G
    __shared__ float As[Qq][Nn];        // A
    __shared__ float Hs[Nn][Nn];        // H (then its Cholesky factor, lower tri)
    __shared__ float Xs[Nn][Qq + 2];    // [rhs_z | A^T], 17 cols used (stride 18)
    __shared__ float Ss[Qq][Qq + 1];    // Schur complement S = A H^-1 A^T
    __shared__ float pv[Nn], hv[Mm], zv[Nn], sv[Mm], lamv[Mm], dv[Mm];
    __shared__ float r1v[Nn], r3v[Mm], wv[Mm], dzv[Nn], dsv[Mm], dlamv[Mm];
    __shared__ float nuv[Qq], r4v[Qq], dnuv[Qq], rnv[Qq], bvv_[Qq];
    __shared__ float red[128];
    __shared__ float xrow[DIN];
    __shared__ float scal[2];

    const int tid = threadIdx.x;
    const int b   = blockIdx.x;

    // ---- Load shared matrices (L2-resident broadcasts), stage L into Hs ----
    for (int t = tid; t < Nn * Nn; t += 128) {
        Gs[t >> 6][t & 63] = Gg[t];
        Hs[t >> 6][t & 63] = Lr[t];     // temp: L_raw
    }
    for (int t = tid; t < Qq * Nn; t += 128) As[t >> 6][t & 63] = Aa[t];
    if (tid < Qq)  bvv_[tid] = beq[tid];
    if (tid < DIN) xrow[tid] = x[b * DIN + tid];
    __syncthreads();

    // ---- Q = tril(L) tril(L)^T + eps I ----
    for (int t = tid; t < Nn * Nn; t += 128) {
        int i = t >> 6, j = t & 63;
        int km = (i < j) ? i : j;
        float acc = (i == j) ? EPSQ : 0.f;
        for (int k = 0; k <= km; ++k) acc += Hs[i][k] * Hs[j][k];
        Qs[i][j] = acc;
    }
    // ---- p = relu(x W_in^T + b_in); h = G z_p + s_p; IPM init ----
    if (tid < Nn) {
        float acc = bin[tid];
        for (int k = 0; k < DIN; ++k) acc += xrow[k] * Win[tid * DIN + k];
        pv[tid] = acc > 0.f ? acc : 0.f;
        float hh = sp[tid];
        for (int k = 0; k < Nn; ++k) hh += Gs[tid][k] * zp[k];
        hv[tid]  = hh;
        zv[tid]  = 0.f;
        sv[tid]  = 1.f;
        lamv[tid] = 1.f;
    }
    if (tid < Qq) nuv[tid] = 0.f;
    __syncthreads();

    // ---- 25 interior-point iterations ----
    for (int it = 0; it < NIT; ++it) {
        // Residuals r1, r3, r4
        if (tid < Nn) {
            float a1 = pv[tid];
            for (int j = 0; j < Nn; ++j)  a1 += Qs[tid][j] * zv[j];   // z@Q^T (Q sym)
            for (int q = 0; q < Qq; ++q)  a1 += As[q][tid] * nuv[q];  // nu@A
            for (int m = 0; m < Mm; ++m)  a1 += Gs[m][tid] * lamv[m]; // lam@G
            r1v[tid] = a1;
            float a3 = sv[tid] - hv[tid];
            for (int j = 0; j < Nn; ++j)  a3 += Gs[tid][j] * zv[j];
            r3v[tid] = a3;
        }
        if (tid < Qq) {
            float a4 = -bvv_[tid];
            for (int j = 0; j < Nn; ++j) a4 += As[tid][j] * zv[j];
            r4v[tid] = a4;
        }
        // mu = mean(s*lam)
        red[tid] = (tid < Mm) ? sv[tid] * lamv[tid] : 0.f;
        __syncthreads();
        for (int off = 64; off > 0; off >>= 1) {
            if (tid < off) red[tid] += red[tid + off];
            __syncthreads();
        }
        if (tid == 0) scal[0] = red[0] / (float)Mm;
        __syncthreads();
        const float mu = scal[0];

        if (tid < Mm) {
            float dd = lamv[tid] / sv[tid];
            dv[tid] = dd;
            wv[tid] = SIG * mu / sv[tid] - lamv[tid] + dd * r3v[tid];
        }
        __syncthreads();

        // ---- H = Q + G^T diag(d) G via V_WMMA_F32_16X16X4_F32 ----
        // 16 tiles of 16x16; wave w owns row-block w (tiles (w, 0..3)); K=64 as 16 x (K=4).
        {
            const int lane = tid & 31, wave = tid >> 5;
            const int hh = lane >> 4, l = lane & 15;
            const int iB = wave << 4;
            for (int t4 = 0; t4 < 4; ++t4) {
                const int jB = t4 << 4;
                v8f acc = {0.f, 0.f, 0.f, 0.f, 0.f, 0.f, 0.f, 0.f};
                for (int kc = 0; kc < 16; ++kc) {
                    const int ka = (kc << 2) + (hh << 1);
                    v2f af, bf;
                    // A[i,k] = d[k]*G[k][iB+i]  (diag(d) folded into A fragment)
                    af.x = dv[ka]     * Gs[ka][iB + l];
                    af.y = dv[ka + 1] * Gs[ka + 1][iB + l];
                    // B[k,j] = G[k][jB+j]
                    bf.x = Gs[ka][jB + l];
                    bf.y = Gs[ka + 1][jB + l];
                    acc = __builtin_amdgcn_wmma_f32_16x16x4_f32(
                        false, af, false, bf, (short)0, acc, false, false);
                }
                for (int r = 0; r < 8; ++r) {
                    const int row = iB + (hh << 3) + r;  // C layout: VGPR r -> M=r / M=8+r
                    Hs[row][jB + l] = acc[r] + Qs[row][jB + l];
                }
            }
        }
        // RHS block: col 0 = rhs_z, cols 1..16 = A^T
        if (tid < Nn) {
            float rz = -r1v[tid];
            for (int m = 0; m < Mm; ++m) rz -= wv[m] * Gs[m][tid];
            Xs[tid][0] = rz;
        }
        for (int t = tid; t < Qq * Nn; t += 128) {
            int q = t >> 6, i = t & 63;
            Xs[i][1 + q] = As[q][i];
        }
        __syncthreads();

        // ---- Cholesky of H (lower triangle), barrier-synchronized ----
        for (int k = 0; k < Nn; ++k) {
            if (tid == 0) Hs[k][k] = sqrtf(Hs[k][k]);
            __syncthreads();
            for (int j = k + 1 + tid; j < Nn; j += 128) Hs[j][k] /= Hs[k][k];
            __syncthreads();
            const int rem = Nn - 1 - k;
            for (int t = tid; t < rem * rem; t += 128) {
                int i = k + 1 + t / rem, j = k + 1 + t % rem;
                if (j <= i) Hs[i][j] -= Hs[i][k] * Hs[j][k];
            }
            __syncthreads();
        }
        // ---- Forward solve L Y = RHS (17 columns) ----
        for (int k = 0; k < Nn; ++k) {
            if (tid < 17) Xs[k][tid] /= Hs[k][k];
            __syncthreads();
            for (int t = tid; t < (Nn - 1 - k) * 17; t += 128) {
                int j = k + 1 + t / 17, c = t % 17;
                Xs[j][c] -= Hs[j][k] * Xs[k][c];
            }
            __syncthreads();
        }
        // ---- Back solve L^T X = Y ----
        for (int k = Nn - 1; k >= 0; --k) {
            if (tid < 17) Xs[k][tid] /= Hs[k][k];
            __syncthreads();
            for (int t = tid; t < k * 17; t += 128) {
                int j = t / 17, c = t % 17;
                Xs[j][c] -= Hs[k][j] * Xs[k][c];
            }
            __syncthreads();
        }
        // ---- Schur: S = A (H^-1 A^T); rhs_nu = A (H^-1 rhs_z) + r4 ----
        for (int t = tid; t < Qq * Qq; t += 128) {
            int p = t >> 4, q = t & 15;
            float a = 0.f;
            for (int n = 0; n < Nn; ++n) a += As[p][n] * Xs[n][1 + q];
            Ss[p][q] = a;
        }
        if (tid < Qq) {
            float a = r4v[tid];
            for (int n = 0; n < Nn; ++n) a += As[tid][n] * Xs[n][0];
            rnv[tid] = a;
        }
        __syncthreads();
        // Tiny 16x16 SPD solve, sequential on one lane (negligible cost)
        if (tid == 0) {
            for (int k = 0; k < Qq; ++k) {
                float dk = sqrtf(Ss[k][k]); Ss[k][k] = dk;
                for (int j = k + 1; j < Qq; ++j) Ss[j][k] /= dk;
                for (int i = k + 1; i < Qq; ++i)
                    for (int j = k + 1; j <= i; ++j) Ss[i][j] -= Ss[i][k] * Ss[j][k];
            }
            for (int k = 0; k < Qq; ++k) {
                float a = rnv[k];
                for (int j = 0; j < k; ++j) a -= Ss[k][j] * dnuv[j];
                dnuv[k] = a / Ss[k][k];
            }
            for (int k = Qq - 1; k >= 0; --k) {
                float a = dnuv[k];
                for (int j = k + 1; j < Qq; ++j) a -= Ss[j][k] * dnuv[j];
                dnuv[k] = a / Ss[k][k];
            }
        }
        __syncthreads();
        // ---- dz, ds, dlam ----
        if (tid < Nn) {
            float a = Xs[tid][0];
            for (int q = 0; q < Qq; ++q) a -= Xs[tid][1 + q] * dnuv[q];
            dzv[tid] = a;
        }
        __syncthreads();
        if (tid < Mm) {
            float a = -r3v[tid];
            for (int n = 0; n < Nn; ++n) a -= Gs[tid][n] * dzv[n];
            dsv[tid] = a;
            dlamv[tid] = (SIG * mu - sv[tid] * lamv[tid] - lamv[tid] * a) / sv[tid];
        }
        __syncthreads();
        // ---- step length: alpha = min(1, 0.99*min(maxstep(s,ds), maxstep(lam,dlam))) ----
        float ratio = 1e30f;
        if (tid < Mm) {
            float dd = dsv[tid];
            if (dd < 0.f) ratio = -sv[tid] / dd;
        } else {
            int m = tid - Mm;
            float dd = dlamv[m];
            if (dd < 0.f) ratio = -lamv[m] / dd;
        }
        red[tid] = ratio;
        __syncthreads();
        for (int off = 64; off > 0; off >>= 1) {
            if (tid < off) red[tid] = fminf(red[tid], red[tid + off]);
            __syncthreads();
        }
        const float alpha = fminf(1.f, 0.99f * red[0]);
        if (tid < Nn) zv[tid] += alpha * dzv[tid];
        if (tid < Mm) { sv[tid] += alpha * dsv[tid]; lamv[tid] += alpha * dlamv[tid]; }
        if (tid < Qq) nuv[tid] += alpha * dnuv[tid];
        __syncthreads();
    }

    // ---- out = sigmoid(u W_out^T + b_out) ----
    if (tid < DOUT) {
        float a = bout[tid];
        for (int n = 0; n < Nn; ++n) a += zv[n] * Wout[tid * Nn + n];
        out[b * DOUT + tid] = 1.f / (1.f + __expf(-a));
    }
}

extern "C" void kernel_launch(void* const* d_in, const int* in_sizes, int n_in,
                              void* d_out, int out_size, void* d_ws, size_t ws_size,
                              hipStream_t stream) {
    (void)in_sizes; (void)n_in; (void)out_size; (void)d_ws; (void)ws_size;
    const float* x    = (const float*)d_in[0];
    const float* Win  = (const float*)d_in[1];
    const float* bin  = (const float*)d_in[2];
    const float* Lr   = (const float*)d_in[3];
    const float* Gg   = (const float*)d_in[4];
    const float* zp   = (const float*)d_in[5];
    const float* sp   = (const float*)d_in[6];
    const float* Aa   = (const float*)d_in[7];
    const float* beq  = (const float*)d_in[8];
    const float* Wout = (const float*)d_in[9];
    const float* bout = (const float*)d_in[10];
    optnet_ipm_kernel<<<Bn, 128, 0, stream>>>(x, Win, bin, Lr, Gg, zp, sp, Aa, beq,
                                              Wout, bout, (float*)d_out);
}